// SinkhornKnopp_32452772888869
// MI455X (gfx1250) — compile-verified
//
#include <hip/hip_runtime.h>

// Sinkhorn-Knopp, fully factored into scale vectors:
//   E = exp(logits)  (never materialized; recomputed per streaming pass)
//   r1[p] = 1/(colsumE[p]*K)
//   pass i: V[b] = sum_p E[b,p]*r_i[p]; c_i[b] = 1/(V[b]*B);
//           W[p] += E[b,p]*c_i[b];  r_{i+1}[p] = 1/(W[p]*K)
//   out[b,p] = E[b,p]*r3[p]*c3[b]*B
#define ROWS 32768
#define COLS 3000
#define C4   750            // COLS / 4 (row pitch is 16B-aligned: 12000B)
#define BLK  256            // 8 waves (wave32)
#define RPW  32             // rows per workgroup
#define NWG  (ROWS / RPW)   // 1024

// ---------- CDNA5 async global->LDS streaming path ----------
#if defined(__gfx1250__) && \
    __has_builtin(__builtin_amdgcn_global_load_async_to_lds_b128) && \
    __has_builtin(__builtin_amdgcn_s_wait_asynccnt)
#define USE_ASYNC 1
#else
#define USE_ASYNC 0
#endif

typedef int v4i __attribute__((vector_size(16)));
#if USE_ASYNC
typedef __attribute__((address_space(1))) v4i* g_v4i_ptr;
typedef __attribute__((address_space(3))) v4i* l_v4i_ptr;
#endif

__device__ __forceinline__ void async_copy_b128(const float4* g, float4* l) {
#if USE_ASYNC
  __builtin_amdgcn_global_load_async_to_lds_b128(
      (g_v4i_ptr)(v4i*)const_cast<float4*>(g),
      (l_v4i_ptr)(v4i*)l,
      0, 0);
#else
  *l = *g;   // portable fallback (sync copy)
#endif
}
__device__ __forceinline__ void wait_async_le3() {
#if USE_ASYNC
  __builtin_amdgcn_s_wait_asynccnt(3);
#endif
}
__device__ __forceinline__ void wait_async_le0() {
#if USE_ASYNC
  __builtin_amdgcn_s_wait_asynccnt(0);
#endif
}

__device__ __forceinline__ float wave_reduce(float v) {
  #pragma unroll
  for (int m = 16; m >= 1; m >>= 1) v += __shfl_xor(v, m, 32);
  return v;
}

// ---------- pass 0: unweighted column sums of E ----------
__global__ __launch_bounds__(BLK) void sk_colsum(const float* __restrict__ logits,
                                                 float* __restrict__ W) {
  int col = blockIdx.x * BLK + threadIdx.x;
  if (col >= COLS) return;
  const float* p = logits + (size_t)blockIdx.y * 128 * COLS + col;
  float s = 0.f;
  #pragma unroll 4
  for (int i = 0; i < 128; ++i) s += __expf(p[(size_t)i * COLS]);
  atomicAdd(&W[col], s);
}

// ---------- tiny kernels: r = 1/(W*K); W := 0 ----------
__global__ void sk_zero(float* __restrict__ W) {
  int i = blockIdx.x * blockDim.x + threadIdx.x;
  if (i < COLS) W[i] = 0.f;
}
__global__ void sk_make_r(float* __restrict__ W, float* __restrict__ r) {
  int i = blockIdx.x * blockDim.x + threadIdx.x;
  if (i < COLS) {
    r[i] = 1.0f / (W[i] * (float)ROWS);
    W[i] = 0.f;  // ready for next accumulation pass
  }
}

// ---------- fused col-step(i) + row-step(i+1): stream E, double-buffered LDS ----------
__global__ __launch_bounds__(BLK) void sk_mid(const float* __restrict__ logits,
                                              const float* __restrict__ r,
                                              float* __restrict__ Wnext) {
  __shared__ float4 buf[2][C4];   // raw logits rows (async-filled)
  __shared__ float4 rs[C4];       // r vector
  __shared__ float4 acc[C4];      // local W accumulation (thread-owned slots)
  __shared__ float  red[BLK / 32];
  __shared__ float  cbc;

  const int tid  = threadIdx.x;
  const int row0 = blockIdx.x * RPW;
  const float4* g  = (const float4*)logits;
  const float4* gr = (const float4*)r;

  for (int j = tid; j < C4; j += BLK) {
    rs[j] = gr[j];
    float4 z; z.x = z.y = z.z = z.w = 0.f;
    acc[j] = z;
  }
  {  // prefetch row 0 (3 async b128 per wave)
    const float4* grow = g + (size_t)row0 * C4;
    for (int j = tid; j < C4; j += BLK) async_copy_b128(grow + j, &buf[0][j]);
  }
  __syncthreads();

  float4 ev[3];
  for (int i = 0; i < RPW; ++i) {
    const int cur = i & 1;
    if (i + 1 < RPW) {  // prefetch row i+1 into the other buffer
      const float4* grow = g + (size_t)(row0 + i + 1) * C4;
      for (int j = tid; j < C4; j += BLK) async_copy_b128(grow + j, &buf[cur ^ 1][j]);
      wait_async_le3();  // in-order completion => row i's 3 ops are done
    } else {
      wait_async_le0();
    }
    __syncthreads();  // row i visible in LDS to all waves

    // phase 1: V = sum_p exp(x)*r[p]; keep e in registers
    float v = 0.f; int n = 0;
    for (int j = tid; j < C4; j += BLK) {
      float4 x = buf[cur][j];
      float4 rr = rs[j];
      float4 e;
      e.x = __expf(x.x); e.y = __expf(x.y); e.z = __expf(x.z); e.w = __expf(x.w);
      ev[n++] = e;
      v = fmaf(e.x, rr.x, v); v = fmaf(e.y, rr.y, v);
      v = fmaf(e.z, rr.z, v); v = fmaf(e.w, rr.w, v);
    }
    v = wave_reduce(v);
    if ((tid & 31) == 0) red[tid >> 5] = v;
    __syncthreads();
    if (tid == 0) {
      float V = 0.f;
      #pragma unroll
      for (int w = 0; w < BLK / 32; ++w) V += red[w];
      cbc = 1.0f / (V * (float)COLS);
    }
    __syncthreads();
    const float c = cbc;

    // phase 2: acc[p] += e*c  (thread-owned p slots -> race-free)
    n = 0;
    for (int j = tid; j < C4; j += BLK) {
      float4 a = acc[j];
      float4 e = ev[n++];
      a.x = fmaf(e.x, c, a.x); a.y = fmaf(e.y, c, a.y);
      a.z = fmaf(e.z, c, a.z); a.w = fmaf(e.w, c, a.w);
      acc[j] = a;
    }
    __syncthreads();  // all reads of buf[cur] done before it is refilled
  }

  for (int j = tid; j < C4; j += BLK) {  // flush local W to global
    float4 a = acc[j];
    atomicAdd(&Wnext[4 * j + 0], a.x);
    atomicAdd(&Wnext[4 * j + 1], a.y);
    atomicAdd(&Wnext[4 * j + 2], a.z);
    atomicAdd(&Wnext[4 * j + 3], a.w);
  }
}

// ---------- final pass: c3 per row, write out = E*r3*c3*B ----------
__global__ __launch_bounds__(BLK) void sk_final(const float* __restrict__ logits,
                                                const float* __restrict__ r,
                                                float* __restrict__ out) {
  __shared__ float4 buf[2][C4];
  __shared__ float4 rs[C4];
  __shared__ float  red[BLK / 32];
  __shared__ float  cbc;

  const int tid  = threadIdx.x;
  const int row0 = blockIdx.x * RPW;
  const float4* g  = (const float4*)logits;
  const float4* gr = (const float4*)r;
  float4* o4 = (float4*)out;

  for (int j = tid; j < C4; j += BLK) rs[j] = gr[j];
  {
    const float4* grow = g + (size_t)row0 * C4;
    for (int j = tid; j < C4; j += BLK) async_copy_b128(grow + j, &buf[0][j]);
  }
  __syncthreads();

  float4 ev[3];
  for (int i = 0; i < RPW; ++i) {
    const int cur = i & 1;
    const int row = row0 + i;
    if (i + 1 < RPW) {
      const float4* grow = g + (size_t)(row + 1) * C4;
      for (int j = tid; j < C4; j += BLK) async_copy_b128(grow + j, &buf[cur ^ 1][j]);
      wait_async_le3();
    } else {
      wait_async_le0();
    }
    __syncthreads();

    float v = 0.f; int n = 0;
    for (int j = tid; j < C4; j += BLK) {
      float4 x = buf[cur][j];
      float4 rr = rs[j];
      float4 e;
      e.x = __expf(x.x); e.y = __expf(x.y); e.z = __expf(x.z); e.w = __expf(x.w);
      ev[n++] = e;
      v = fmaf(e.x, rr.x, v); v = fmaf(e.y, rr.y, v);
      v = fmaf(e.z, rr.z, v); v = fmaf(e.w, rr.w, v);
    }
    v = wave_reduce(v);
    if ((tid & 31) == 0) red[tid >> 5] = v;
    __syncthreads();
    if (tid == 0) {
      float V = 0.f;
      #pragma unroll
      for (int w = 0; w < BLK / 32; ++w) V += red[w];
      cbc = 1.0f / (V * (float)COLS);
    }
    __syncthreads();
    const float cB = cbc * (float)COLS;  // fold final *B

    n = 0;
    for (int j = tid; j < C4; j += BLK) {
      float4 e = ev[n++];
      float4 rr = rs[j];
      float4 w;
      w.x = e.x * rr.x * cB; w.y = e.y * rr.y * cB;
      w.z = e.z * rr.z * cB; w.w = e.w * rr.w * cB;
      o4[(size_t)row * C4 + j] = w;
    }
    __syncthreads();
  }
}

extern "C" void kernel_launch(void* const* d_in, const int* in_sizes, int n_in,
                              void* d_out, int out_size, void* d_ws, size_t ws_size,
                              hipStream_t stream) {
  (void)in_sizes; (void)n_in; (void)out_size; (void)ws_size;
  const float* logits = (const float*)d_in[0];
  float* out = (float*)d_out;
  float* W = (float*)d_ws;                          // 3000 f32
  float* r = (float*)((char*)d_ws + 16384);         // 3000 f32, 16KB offset

  const int tb = (COLS + 255) / 256;  // 12
  sk_zero<<<tb, 256, 0, stream>>>(W);
  sk_colsum<<<dim3(tb, ROWS / 128), BLK, 0, stream>>>(logits, W);  // W = colsumE
  sk_make_r<<<tb, 256, 0, stream>>>(W, r);                         // r1; W=0
  sk_mid<<<NWG, BLK, 0, stream>>>(logits, r, W);                   // c1, W2
  sk_make_r<<<tb, 256, 0, stream>>>(W, r);                         // r2; W=0
  sk_mid<<<NWG, BLK, 0, stream>>>(logits, r, W);                   // c2, W3
  sk_make_r<<<tb, 256, 0, stream>>>(W, r);                         // r3
  sk_final<<<NWG, BLK, 0, stream>>>(logits, r, out);               // c3, out
}